// DWSpiralDeblock_30322469110336
// MI455X (gfx1250) — compile-verified
//
#include <hip/hip_runtime.h>
#include <hip/hip_bf16.h>

typedef __attribute__((ext_vector_type(2))) float v2f;
typedef __attribute__((ext_vector_type(8))) float v8f;

#define B_     8
#define N_IN_  25000
#define N_OUT_ 50000
#define S_     9
#define C_     64
#define E_     (3 * N_OUT_)

// LDS row padding to avoid stride-64 bank conflicts (64 banks, 4B each)
#define PAD_   68

// ---------------------------------------------------------------------------
// Stage 1: scatter-add   pooled[b, row[e], c] += x[b, col[e], c] * val[e]
// One thread per (e, c); loops over batch. Coalesced 256B per edge.
// ---------------------------------------------------------------------------
__global__ __launch_bounds__(256) void scatter_kernel(
    const float* __restrict__ x,
    const int*   __restrict__ trans_row,
    const int*   __restrict__ trans_col,
    const float* __restrict__ trans_val,
    float*       __restrict__ pooled) {
  int t = blockIdx.x * 256 + threadIdx.x;
  if (t >= E_ * C_) return;
  int e = t >> 6;
  int c = t & 63;
  int row   = trans_row[e];
  int col   = trans_col[e];
  float val = trans_val[e];
  const float* xp = x      + (size_t)col * C_ + c;
  float*       pp = pooled + (size_t)row * C_ + c;
#pragma unroll
  for (int b = 0; b < B_; ++b) {
    atomicAdd(pp + (size_t)b * N_OUT_ * C_,
              xp[(size_t)b * N_IN_ * C_] * val);
  }
}

// ---------------------------------------------------------------------------
// Stage 2: fused gather + depthwise + pointwise GEMM (fp32 WMMA) + ReLU
// Each wave32 owns one (b, 16-row) tile of the output.
//   dw[r, c] = sum_s pooled[b, indices[n0+r, s], c] * dw_weight[c, s]
//   out[r, o] = relu( sum_c dw[r, c] * pw_weight[o, c] )   via V_WMMA_F32_16X16X4_F32
// ---------------------------------------------------------------------------
__global__ __launch_bounds__(256) void fused_dw_pw_kernel(
    const float* __restrict__ pooled,
    const int*   __restrict__ indices,
    const float* __restrict__ dw_w,   // (C, S)
    const float* __restrict__ pw_w,   // (C_OUT, C_IN)
    float*       __restrict__ out) {
  __shared__ float s_pw[C_ * PAD_];        // pw_weight, padded rows
  __shared__ float s_dww[C_ * S_];         // dw_weight
  __shared__ float s_dw[8][16 * PAD_];     // per-wave 16x64 dw tile, padded

  const int tid  = threadIdx.x;
  const int lane = tid & 31;
  const int wave = tid >> 5;

  // Cooperative staging of the (small) weights into LDS.
  for (int i = tid; i < C_ * C_; i += 256) {
    int o = i >> 6, c = i & 63;
    s_pw[o * PAD_ + c] = pw_w[i];
  }
  for (int i = tid; i < C_ * S_; i += 256) s_dww[i] = dw_w[i];
  __syncthreads();

  // Tile assignment: 8 waves/block, 3125 tiles per batch element.
  const int tile = blockIdx.x * 8 + wave;        // [0, 25000)
  const int b    = tile / (N_OUT_ / 16);
  const int n0   = (tile % (N_OUT_ / 16)) * 16;
  const float* pooledB = pooled + (size_t)b * N_OUT_ * C_;

  // ---- gather + depthwise: build 16x64 dw tile in LDS ----
  // lanes cover channels {lane, lane+32} for each of the 16 rows
  for (int r = 0; r < 16; ++r) {
    const int n = n0 + r;
    float a0 = 0.f, a1 = 0.f;
#pragma unroll
    for (int s = 0; s < S_; ++s) {
      const int idx = indices[n * S_ + s];            // wave-uniform
      const float* src = pooledB + (size_t)idx * C_;
      a0 += src[lane]      * s_dww[lane * S_ + s];
      a1 += src[lane + 32] * s_dww[(lane + 32) * S_ + s];
    }
    s_dw[wave][r * PAD_ + lane]      = a0;
    s_dw[wave][r * PAD_ + lane + 32] = a1;
  }
  __syncthreads();

  // ---- pointwise GEMM: D(16x64) = dw(16x64) x pw^T(64x64), K in steps of 4 ----
  v8f acc[4] = {};                         // 4 n-tiles of 16x16 fp32 accum
  const int row  = lane & 15;
  const int koff = (lane < 16) ? 0 : 2;    // f32 16x4 A layout: hi half-wave holds K+2,K+3

#pragma unroll
  for (int kk = 0; kk < C_; kk += 4) {
    v2f a;
    a.x = s_dw[wave][row * PAD_ + kk + koff];
    a.y = s_dw[wave][row * PAD_ + kk + koff + 1];
#pragma unroll
    for (int nt = 0; nt < 4; ++nt) {
      const int o = nt * 16 + row;         // B[k][n] = pw_w[o=n][c=k]
      v2f bf;
      bf.x = s_pw[o * PAD_ + kk + koff];
      bf.y = s_pw[o * PAD_ + kk + koff + 1];
      acc[nt] = __builtin_amdgcn_wmma_f32_16x16x4_f32(
          /*neg_a=*/false, a, /*neg_b=*/false, bf,
          /*c_mod=*/(short)0, acc[nt], /*reuse_a=*/false, /*reuse_b=*/false);
    }
  }

  // ---- ReLU + store. C/D layout: VGPR j -> M = j + (lane<16 ? 0 : 8), N = lane&15 ----
  const int mbase = (lane < 16) ? 0 : 8;
  float* outB = out + ((size_t)b * N_OUT_ + n0) * C_;
#pragma unroll
  for (int nt = 0; nt < 4; ++nt) {
    const int o = nt * 16 + row;
#pragma unroll
    for (int j = 0; j < 8; ++j) {
      float v = acc[nt][j];
      outB[(size_t)(mbase + j) * C_ + o] = v > 0.f ? v : 0.f;
    }
  }
}

// ---------------------------------------------------------------------------
extern "C" void kernel_launch(void* const* d_in, const int* in_sizes, int n_in,
                              void* d_out, int out_size, void* d_ws, size_t ws_size,
                              hipStream_t stream) {
  const float* x         = (const float*)d_in[0];
  const int*   trans_row = (const int*)  d_in[1];
  const int*   trans_col = (const int*)  d_in[2];
  const float* trans_val = (const float*)d_in[3];
  const int*   indices   = (const int*)  d_in[4];
  const float* dw_w      = (const float*)d_in[5];
  const float* pw_w      = (const float*)d_in[6];
  float*       out       = (float*)d_out;

  float* pooled = (float*)d_ws;  // (B, N_OUT, C) fp32 = 102.4 MB scratch
  const size_t pooled_bytes = (size_t)B_ * N_OUT_ * C_ * sizeof(float);
  hipMemsetAsync(pooled, 0, pooled_bytes, stream);

  const int scatter_work = E_ * C_;
  scatter_kernel<<<(scatter_work + 255) / 256, 256, 0, stream>>>(
      x, trans_row, trans_col, trans_val, pooled);

  // 25,000 tiles total, 8 waves (tiles) per 256-thread block
  fused_dw_pw_kernel<<<(B_ * (N_OUT_ / 16)) / 8, 256, 0, stream>>>(
      pooled, indices, dw_w, pw_w, out);
}